// PillarFeatureNet_63977832841498
// MI455X (gfx1250) — compile-verified
//
#include <hip/hip_runtime.h>
#include <math.h>

// ---------------- constants (match reference) ----------------
static constexpr int   NPTS  = 500000;
static constexpr int   GXC   = 512;
static constexpr int   GYC   = 512;
static constexpr int   SSEG  = 2 * GXC * GYC;        // 524288 segments
static constexpr float BN_EPS = 0.001f;
static constexpr float PC_MINX = -51.2f, PC_MINY = -51.2f;
static constexpr float VOXX = 0.2f, VOXY = 0.2f;

typedef float v2f __attribute__((ext_vector_type(2)));
typedef float v8f __attribute__((ext_vector_type(8)));

__device__ __forceinline__ unsigned enc_ord(float f) {
  unsigned u = __float_as_uint(f);
  return (u & 0x80000000u) ? ~u : (u | 0x80000000u);
}
__device__ __forceinline__ float dec_ord(unsigned e) {
  unsigned u = (e & 0x80000000u) ? (e & 0x7fffffffu) : ~e;
  return __uint_as_float(u);
}

// ---------------- generic zero (uint4 = 16B stores) ----------------
__global__ void k_zero4(uint4* __restrict__ p, size_t n4) {
  size_t i = (size_t)blockIdx.x * blockDim.x + threadIdx.x;
  if (i < n4) p[i] = make_uint4(0u, 0u, 0u, 0u);
}

// ---------------- voxelize: seg ids, counts, xyz sums ----------------
__global__ void k_voxel(const float* __restrict__ pts, int* __restrict__ seg,
                        unsigned* __restrict__ cnt, float* __restrict__ sums) {
  int i = blockIdx.x * blockDim.x + threadIdx.x;
  if (i >= NPTS) return;
  const float* p = pts + (size_t)i * 6;
  float b = p[0], x = p[1], y = p[2], z = p[3];
  int ii = (int)floorf((x - PC_MINX) / VOXX);
  int jj = (int)floorf((y - PC_MINY) / VOXY);
  ii = min(max(ii, 0), GXC - 1);
  jj = min(max(jj, 0), GYC - 1);
  int s = (((int)b) * GXC + ii) * GYC + jj;
  seg[i] = s;
  atomicAdd(&cnt[s], 1u);
  atomicAdd(&sums[s], x);
  atomicAdd(&sums[SSEG + s], y);
  atomicAdd(&sums[2 * SSEG + s], z);
}

// ---------------- sums -> means (in place) ----------------
__global__ void k_mean(const unsigned* __restrict__ cnt, float* __restrict__ sums) {
  int s = blockIdx.x * blockDim.x + threadIdx.x;
  if (s >= SSEG) return;
  float inv = 1.0f / fmaxf((float)cnt[s], 1.0f);
  sums[s] *= inv;
  sums[SSEG + s] *= inv;
  sums[2 * SSEG + s] *= inv;
}

// ---------------- layer 0: features, 10x32 GEMM, BN stat partials ----------------
static constexpr int NB0 = 1024;
__global__ __launch_bounds__(256) void k_l0(const float* __restrict__ pts,
                                            const int* __restrict__ seg,
                                            const float* __restrict__ means,
                                            const float* __restrict__ W0,
                                            float* __restrict__ x0,
                                            float* __restrict__ part) {
  __shared__ float Wl[320];
  __shared__ float red[64];
  int tid = threadIdx.x;
  for (int t = tid; t < 320; t += 256) Wl[t] = W0[t];
  if (tid < 64) red[tid] = 0.0f;
  __syncthreads();

  float acc[32], acc2[32];
#pragma unroll
  for (int c = 0; c < 32; ++c) { acc[c] = 0.0f; acc2[c] = 0.0f; }

  for (int i = blockIdx.x * 256 + tid; i < NPTS; i += gridDim.x * 256) {
    const float* p = pts + (size_t)i * 6;
    float x = p[1], y = p[2], z = p[3], e1 = p[4], e2 = p[5];
    int ii = min(max((int)floorf((x - PC_MINX) / VOXX), 0), GXC - 1);
    int jj = min(max((int)floorf((y - PC_MINY) / VOXY), 0), GYC - 1);
    int s = seg[i];
    float mx = means[s], my = means[SSEG + s], mz = means[2 * SSEG + s];
    float f[10];
    f[0] = x; f[1] = y; f[2] = z; f[3] = e1; f[4] = e2;
    f[5] = x - mx; f[6] = y - my; f[7] = z - mz;
    f[8] = x - ((float)ii * VOXX + 0.5f * VOXX + PC_MINX);
    f[9] = y - ((float)jj * VOXY + 0.5f * VOXY + PC_MINY);
#pragma unroll
    for (int c = 0; c < 32; ++c) {
      float v = 0.0f;
#pragma unroll
      for (int k = 0; k < 10; ++k) v = fmaf(f[k], Wl[k * 32 + c], v);
      x0[(size_t)i * 32 + c] = v;
      acc[c] += v;
      acc2[c] += v * v;
    }
  }
#pragma unroll
  for (int c = 0; c < 32; ++c) {
    atomicAdd(&red[c], acc[c]);
    atomicAdd(&red[32 + c], acc2[c]);
  }
  __syncthreads();
  if (tid < 64) part[(size_t)blockIdx.x * 64 + tid] = red[tid];
}

// ---------------- finalize BN0 params ----------------
__global__ void k_bn0(const float* __restrict__ part, const float* __restrict__ gamma,
                      const float* __restrict__ beta, float* __restrict__ bn0) {
  __shared__ float tot[64];
  int t = threadIdx.x;  // 64 threads
  float s = 0.0f;
  for (int b = 0; b < NB0; ++b) s += part[(size_t)b * 64 + t];
  tot[t] = s;
  __syncthreads();
  if (t < 32) {
    float mu  = tot[t] * (1.0f / (float)NPTS);
    float var = tot[32 + t] * (1.0f / (float)NPTS) - mu * mu;
    float A = rsqrtf(var + BN_EPS) * gamma[t];
    bn0[t] = A;
    bn0[32 + t] = beta[t] - mu * A;
  }
}

// ---------------- apply BN0 + PReLU, segment atomics (layer 0) ----------------
__global__ __launch_bounds__(256) void k_act0(float* __restrict__ x0,
                                              const int* __restrict__ seg,
                                              const float* __restrict__ bn0,
                                              const float* __restrict__ a0,
                                              unsigned* __restrict__ smax,
                                              float* __restrict__ ssum) {
  int i = blockIdx.x * blockDim.x + threadIdx.x;
  if (i >= NPTS) return;
  int s = seg[i];
#pragma unroll
  for (int c = 0; c < 32; ++c) {
    float v = x0[(size_t)i * 32 + c] * bn0[c] + bn0[32 + c];
    v = v > 0.0f ? v : a0[c] * v;
    x0[(size_t)i * 32 + c] = v;  // in place: x0 becomes post-activation
    atomicAdd(&ssum[(size_t)s * 32 + c], v);
    atomicMax(&smax[(size_t)s * 32 + c], enc_ord(v));
  }
}

// ---------------- hybrid pool (layer 0) ----------------
__global__ void k_hyb0(const unsigned* __restrict__ smax, const float* __restrict__ ssum,
                       const unsigned* __restrict__ cnt, const float* __restrict__ alpha0,
                       float* __restrict__ hyb0) {
  int idx = blockIdx.x * blockDim.x + threadIdx.x;
  if (idx >= SSEG * 32) return;
  int s = idx >> 5;
  unsigned c = cnt[s];
  float al = 1.0f / (1.0f + expf(-alpha0[0]));
  float m  = dec_ord(smax[idx]);
  float sm = ssum[idx] / fmaxf((float)c, 1.0f);
  float h  = al * m + (1.0f - al) * sm;
  hyb0[idx] = (c > 0u) ? h : 0.0f;
}

// =================================================================
// Layer 1: 64x128 GEMM via V_WMMA_F32_16X16X4_F32 (fp32 WMMA).
// Per 128-thread block: 4 waves, each wave owns a 16-point tile.
// A layout (ISA 7.12.2): lanes 0-15 row M, VGPR pair = K{0,1}; lanes
// 16-31 same rows, K{2,3}. B mirrors with N across lanes. C/D v8f:
// VGPR r -> (M=r | r+8, N=lane&15).
// W1 is stored in LDS as K-pairs (float2 {W[2p][n], W[2p+1][n]}) so a
// B fragment is one aligned ds_load_b64. Tile staging (x0a rows +
// seg-gathered hyb0 rows) goes straight to LDS with
// GLOBAL_LOAD_ASYNC_TO_LDS_B128 (ASYNCcnt), bypassing VGPRs.
// =================================================================
static constexpr int NB1 = 512;

__global__ __launch_bounds__(128) void k_gemm1_stats(const float* __restrict__ x0a,
                                                     const float* __restrict__ hyb0,
                                                     const int* __restrict__ seg,
                                                     const float* __restrict__ W1,
                                                     float* __restrict__ part) {
  __shared__ __align__(16) v2f Wp[32 * 128];
  __shared__ __align__(16) float stage[4 * 16 * 64];
  __shared__ int segl[64];
  __shared__ float red[256];
  int tid = threadIdx.x, w = tid >> 5, lane = tid & 31, ln = lane & 15;
  bool hi = (lane >= 16);
  int koff = hi ? 2 : 0;
  int hsel = hi ? 1 : 0;
  for (int t = tid; t < 32 * 128; t += 128) {
    int p = t >> 7, n = t & 127;
    v2f wv;
    wv.x = W1[(2 * p) * 128 + n];
    wv.y = W1[(2 * p + 1) * 128 + n];
    Wp[t] = wv;
  }
  for (int t = tid; t < 256; t += 128) red[t] = 0.0f;

  float sums[8], sqs[8];
#pragma unroll
  for (int nt = 0; nt < 8; ++nt) { sums[nt] = 0.0f; sqs[nt] = 0.0f; }

  unsigned sbase = (unsigned)(uintptr_t)(&stage[0]);
  const int tiles = (NPTS + 15) / 16;
  for (int tb = blockIdx.x * 4; tb < tiles; tb += gridDim.x * 4) {
    __syncthreads();
    if (tid < 64) {
      int gp = (tb + (tid >> 4)) * 16 + (tid & 15);
      segl[tid] = (gp < NPTS) ? seg[gp] : -1;
    }
    __syncthreads();
    bool tail = ((tb + 4) * 16 > NPTS);
    if (tail) {
      for (int t = tid; t < 1024; t += 128)
        *(float4*)(&stage[t * 4]) = make_float4(0.f, 0.f, 0.f, 0.f);
      __syncthreads();
    }
    for (int t = tid; t < 1024; t += 128) {
      int w2 = t >> 8, r = (t >> 4) & 15, c4 = t & 15;
      int gp = (tb + w2) * 16 + r;
      if (gp < NPTS) {
        const float* src;
        if (c4 < 8) src = x0a + (size_t)gp * 32 + c4 * 4;
        else { int sg = segl[w2 * 16 + r];
               src = hyb0 + (size_t)sg * 32 + (c4 - 8) * 4; }
        unsigned ldsoff = sbase + (unsigned)t * 16u;
        unsigned long long ga = (unsigned long long)(uintptr_t)src;
        asm volatile("global_load_async_to_lds_b128 %0, %1, off"
                     :: "v"(ldsoff), "v"(ga) : "memory");
      }
    }
    asm volatile("s_wait_asynccnt 0x0" ::: "memory");
    __syncthreads();

    v2f a[16];
    const float* srow = &stage[w * 1024 + ln * 64];
#pragma unroll
    for (int k = 0; k < 16; ++k) a[k] = *(const v2f*)(srow + 4 * k + koff);

#pragma unroll
    for (int nt = 0; nt < 8; ++nt) {
      v8f cacc = {0.f, 0.f, 0.f, 0.f, 0.f, 0.f, 0.f, 0.f};
      int n = nt * 16 + ln;
#pragma unroll
      for (int k = 0; k < 16; ++k) {
        v2f b = Wp[(2 * k + hsel) * 128 + n];
        cacc = __builtin_amdgcn_wmma_f32_16x16x4_f32(false, a[k], false, b,
                                                     (short)0, cacc, false, false);
      }
#pragma unroll
      for (int r = 0; r < 8; ++r) { float v = cacc[r]; sums[nt] += v; sqs[nt] += v * v; }
    }
  }
  __syncthreads();
#pragma unroll
  for (int nt = 0; nt < 8; ++nt) {
    int ch = nt * 16 + ln;
    atomicAdd(&red[ch], sums[nt]);
    atomicAdd(&red[128 + ch], sqs[nt]);
  }
  __syncthreads();
  for (int t = tid; t < 256; t += 128) part[(size_t)blockIdx.x * 256 + t] = red[t];
}

// ---------------- finalize BN1 params ----------------
__global__ void k_bn1(const float* __restrict__ part, const float* __restrict__ gamma,
                      const float* __restrict__ beta, float* __restrict__ bn1) {
  __shared__ float tot[256];
  int t = threadIdx.x;  // 256 threads
  float s = 0.0f;
  for (int b = 0; b < NB1; ++b) s += part[(size_t)b * 256 + t];
  tot[t] = s;
  __syncthreads();
  if (t < 128) {
    float mu  = tot[t] * (1.0f / (float)NPTS);
    float var = tot[128 + t] * (1.0f / (float)NPTS) - mu * mu;
    float A = rsqrtf(var + BN_EPS) * gamma[t];
    bn1[t] = A;
    bn1[128 + t] = beta[t] - mu * A;
  }
}

// ---------------- layer 1 pass 2: GEMM again + BN + PReLU + scatter ----------------
__global__ __launch_bounds__(128) void k_gemm1_apply(const float* __restrict__ x0a,
                                                     const float* __restrict__ hyb0,
                                                     const int* __restrict__ seg,
                                                     const float* __restrict__ W1,
                                                     const float* __restrict__ bn1,
                                                     const float* __restrict__ a1,
                                                     float* __restrict__ osum,
                                                     unsigned* __restrict__ omax) {
  __shared__ __align__(16) v2f Wp[32 * 128];
  __shared__ __align__(16) float stage[4 * 16 * 64];
  __shared__ int segl[64];
  __shared__ float bnl[384];  // A[128], B[128], slope[128]
  int tid = threadIdx.x, w = tid >> 5, lane = tid & 31, ln = lane & 15;
  bool hi = (lane >= 16);
  int koff = hi ? 2 : 0;
  int hsel = hi ? 1 : 0;
  for (int t = tid; t < 32 * 128; t += 128) {
    int p = t >> 7, n = t & 127;
    v2f wv;
    wv.x = W1[(2 * p) * 128 + n];
    wv.y = W1[(2 * p + 1) * 128 + n];
    Wp[t] = wv;
  }
  for (int t = tid; t < 256; t += 128) bnl[t] = bn1[t];
  if (tid < 128) bnl[256 + tid] = a1[tid];

  unsigned sbase = (unsigned)(uintptr_t)(&stage[0]);
  const int tiles = (NPTS + 15) / 16;
  for (int tb = blockIdx.x * 4; tb < tiles; tb += gridDim.x * 4) {
    __syncthreads();
    if (tid < 64) {
      int gp = (tb + (tid >> 4)) * 16 + (tid & 15);
      segl[tid] = (gp < NPTS) ? seg[gp] : -1;
    }
    __syncthreads();
    bool tail = ((tb + 4) * 16 > NPTS);
    if (tail) {
      for (int t = tid; t < 1024; t += 128)
        *(float4*)(&stage[t * 4]) = make_float4(0.f, 0.f, 0.f, 0.f);
      __syncthreads();
    }
    for (int t = tid; t < 1024; t += 128) {
      int w2 = t >> 8, r = (t >> 4) & 15, c4 = t & 15;
      int gp = (tb + w2) * 16 + r;
      if (gp < NPTS) {
        const float* src;
        if (c4 < 8) src = x0a + (size_t)gp * 32 + c4 * 4;
        else { int sg = segl[w2 * 16 + r];
               src = hyb0 + (size_t)sg * 32 + (c4 - 8) * 4; }
        unsigned ldsoff = sbase + (unsigned)t * 16u;
        unsigned long long ga = (unsigned long long)(uintptr_t)src;
        asm volatile("global_load_async_to_lds_b128 %0, %1, off"
                     :: "v"(ldsoff), "v"(ga) : "memory");
      }
    }
    asm volatile("s_wait_asynccnt 0x0" ::: "memory");
    __syncthreads();

    v2f a[16];
    const float* srow = &stage[w * 1024 + ln * 64];
#pragma unroll
    for (int k = 0; k < 16; ++k) a[k] = *(const v2f*)(srow + 4 * k + koff);

#pragma unroll
    for (int nt = 0; nt < 8; ++nt) {
      v8f cacc = {0.f, 0.f, 0.f, 0.f, 0.f, 0.f, 0.f, 0.f};
      int n = nt * 16 + ln;
#pragma unroll
      for (int k = 0; k < 16; ++k) {
        v2f b = Wp[(2 * k + hsel) * 128 + n];
        cacc = __builtin_amdgcn_wmma_f32_16x16x4_f32(false, a[k], false, b,
                                                     (short)0, cacc, false, false);
      }
      float A = bnl[n], Bb = bnl[128 + n], sl = bnl[256 + n];
#pragma unroll
      for (int r = 0; r < 8; ++r) {
        int M = hi ? (8 + r) : r;
        int sg = segl[w * 16 + M];
        if (sg >= 0) {
          float v = cacc[r] * A + Bb;
          v = v > 0.0f ? v : sl * v;
          atomicAdd(&osum[(size_t)sg * 128 + n], v);
          atomicMax(&omax[(size_t)sg * 128 + n], enc_ord(v));
        }
      }
    }
  }
}

// ---------------- final hybrid pool into d_out (in place over sums) ----------------
__global__ void k_final(float* __restrict__ out, const unsigned* __restrict__ omax,
                        const unsigned* __restrict__ cnt, const float* __restrict__ alpha1) {
  size_t idx = (size_t)blockIdx.x * blockDim.x + threadIdx.x;
  if (idx >= (size_t)SSEG * 128) return;
  int s = (int)(idx >> 7);
  unsigned c = cnt[s];
  float al = 1.0f / (1.0f + expf(-alpha1[0]));
  float h = al * dec_ord(omax[idx]) + (1.0f - al) * (out[idx] / fmaxf((float)c, 1.0f));
  out[idx] = (c > 0u) ? h : 0.0f;
}

static inline int divup(long long a, long long b) { return (int)((a + b - 1) / b); }

extern "C" void kernel_launch(void* const* d_in, const int* in_sizes, int n_in,
                              void* d_out, int out_size, void* d_ws, size_t ws_size,
                              hipStream_t stream) {
  (void)in_sizes; (void)n_in; (void)out_size; (void)ws_size;
  const float* pts    = (const float*)d_in[0];
  const float* W0     = (const float*)d_in[1];
  const float* gamma0 = (const float*)d_in[2];
  const float* beta0  = (const float*)d_in[3];
  const float* a0     = (const float*)d_in[4];
  const float* alpha0 = (const float*)d_in[5];
  const float* W1     = (const float*)d_in[6];
  const float* gamma1 = (const float*)d_in[7];
  const float* beta1  = (const float*)d_in[8];
  const float* a1     = (const float*)d_in[9];
  const float* alpha1 = (const float*)d_in[10];
  float* out = (float*)d_out;

  char* ws = (char*)d_ws;
  size_t off = 0;
  auto take = [&](size_t bytes) -> char* {
    char* r = ws + off;
    off = (off + bytes + 255) & ~(size_t)255;
    return r;
  };
  unsigned* cnt  = (unsigned*)take((size_t)4 * SSEG * 4);      // cnt + 3 mean planes, contiguous
  float*    sums = (float*)(cnt + SSEG);
  int*      seg  = (int*)take((size_t)NPTS * 4);
  float*    x0   = (float*)take((size_t)NPTS * 32 * 4);
  float*    hyb0 = (float*)take((size_t)SSEG * 32 * 4);
  unsigned* omax = (unsigned*)take((size_t)SSEG * 128 * 4);
  float*    part = (float*)take((size_t)1 << 20);
  float*    bn0  = (float*)take(64 * 4);
  float*    bn1  = (float*)take(256 * 4);

  // layer-0 segment scratch lives inside d_out (S*64 floats <= S*128)
  unsigned* smax0 = (unsigned*)out;
  float*    ssum0 = out + (size_t)SSEG * 32;

  // --- voxelize ---
  k_zero4<<<divup(SSEG, 256), 256, 0, stream>>>((uint4*)cnt, (size_t)SSEG);  // 4*S u32
  k_voxel<<<divup(NPTS, 256), 256, 0, stream>>>(pts, seg, cnt, sums);
  k_mean<<<divup(SSEG, 256), 256, 0, stream>>>(cnt, sums);

  // --- layer 0 ---
  k_l0<<<NB0, 256, 0, stream>>>(pts, seg, sums, W0, x0, part);
  k_bn0<<<1, 64, 0, stream>>>(part, gamma0, beta0, bn0);
  k_zero4<<<divup((long long)SSEG * 16, 256), 256, 0, stream>>>((uint4*)out, (size_t)SSEG * 16);
  k_act0<<<divup(NPTS, 256), 256, 0, stream>>>(x0, seg, bn0, a0, smax0, ssum0);
  k_hyb0<<<divup((long long)SSEG * 32, 256), 256, 0, stream>>>(smax0, ssum0, cnt, alpha0, hyb0);

  // --- layer 1 (WMMA f32 16x16x4, async-to-LDS staging) ---
  k_gemm1_stats<<<NB1, 128, 0, stream>>>(x0, hyb0, seg, W1, part);
  k_bn1<<<1, 256, 0, stream>>>(part, gamma1, beta1, bn1);
  k_zero4<<<divup((long long)SSEG * 32, 256), 256, 0, stream>>>((uint4*)out, (size_t)SSEG * 32);
  k_zero4<<<divup((long long)SSEG * 32, 256), 256, 0, stream>>>((uint4*)omax, (size_t)SSEG * 32);
  k_gemm1_apply<<<NB1, 128, 0, stream>>>(x0, hyb0, seg, W1, bn1, a1, out, omax);
  k_final<<<divup((long long)SSEG * 128, 256), 256, 0, stream>>>(out, omax, cnt, alpha1);
}